// GlobalContextBlock_549755814377
// MI455X (gfx1250) — compile-verified
//
#include <hip/hip_runtime.h>
#include <hip/hip_bf16.h>

#define BB 16
#define CC 256
#define CR 16
#define SS 16384          // H*W = 128*128
#define LN_EPS 1e-5f

typedef __attribute__((ext_vector_type(2))) float v2f;
typedef __attribute__((ext_vector_type(8))) float v8f;

// ---------------------------------------------------------------------------
// Kernel 1: logits[b][s] = sum_c x[b][c][s] * w_mask[c] + b_mask
// Pure streaming pass over x (256 MiB). 1024 blocks x 256 threads,
// one pixel per thread, coalesced loads, w_mask staged in LDS.
// ---------------------------------------------------------------------------
__global__ __launch_bounds__(256) void k_logits(const float* __restrict__ x,
                                                const float* __restrict__ w_mask,
                                                const float* __restrict__ b_mask,
                                                float* __restrict__ logits) {
    __shared__ float ws[CC];
    const int tid = threadIdx.x;
    ws[tid] = w_mask[tid];
    __syncthreads();
    const int b = blockIdx.x >> 6;                 // 64 chunks of 256 pixels per batch
    const int s = ((blockIdx.x & 63) << 8) + tid;
    const float* xp = x + (size_t)b * CC * SS + s;
    float acc = 0.f;
    #pragma unroll 8
    for (int c = 0; c < CC; ++c)
        acc = fmaf(xp[(size_t)c * SS], ws[c], acc);
    logits[b * SS + s] = acc + b_mask[0];
}

// ---------------------------------------------------------------------------
// Kernel 2: in-place softmax over S=16384 per batch. One 1024-thread block
// per batch (32 waves). Max -> exp/sum -> normalize, LDS tree reductions.
// ---------------------------------------------------------------------------
__global__ __launch_bounds__(1024) void k_softmax(float* __restrict__ logits) {
    __shared__ float red[1024];
    const int tid = threadIdx.x;
    float* l = logits + blockIdx.x * SS;

    float m = -3.402823466e38f;
    for (int i = 0; i < SS; i += 1024) m = fmaxf(m, l[i + tid]);
    red[tid] = m; __syncthreads();
    for (int off = 512; off > 0; off >>= 1) {
        if (tid < off) red[tid] = fmaxf(red[tid], red[tid + off]);
        __syncthreads();
    }
    const float gmax = red[0];
    __syncthreads();

    float s = 0.f;
    for (int i = 0; i < SS; i += 1024) {
        float e = __expf(l[i + tid] - gmax);
        l[i + tid] = e;
        s += e;
    }
    red[tid] = s; __syncthreads();
    for (int off = 512; off > 0; off >>= 1) {
        if (tid < off) red[tid] += red[tid + off];
        __syncthreads();
    }
    const float inv = 1.f / red[0];
    for (int i = 0; i < SS; i += 1024) l[i + tid] *= inv;
}

// ---------------------------------------------------------------------------
// Kernel 3: context[b][c] = sum_s x[b][c][s] * attn[b][s].
// Second streaming pass over x. Each block handles one batch and 4 channels
// so each attn value is loaded once per 4 channel-MACs (attn is L2-resident
// anyway: 1 MiB total). 1024 blocks x 256 threads, LDS tree reduction.
// ---------------------------------------------------------------------------
__global__ __launch_bounds__(256) void k_context(const float* __restrict__ x,
                                                 const float* __restrict__ attn,
                                                 float* __restrict__ context) {
    __shared__ float r0[256], r1[256], r2[256], r3[256];
    const int tid = threadIdx.x;
    const int b  = blockIdx.x >> 6;
    const int c0 = (blockIdx.x & 63) << 2;
    const float* xp = x + ((size_t)b * CC + c0) * SS;
    const float* ap = attn + b * SS;
    float a0 = 0.f, a1 = 0.f, a2 = 0.f, a3 = 0.f;
    for (int s = tid; s < SS; s += 256) {
        const float a = ap[s];
        a0 = fmaf(xp[s], a, a0);
        a1 = fmaf(xp[(size_t)SS + s], a, a1);
        a2 = fmaf(xp[(size_t)2 * SS + s], a, a2);
        a3 = fmaf(xp[(size_t)3 * SS + s], a, a3);
    }
    r0[tid] = a0; r1[tid] = a1; r2[tid] = a2; r3[tid] = a3;
    __syncthreads();
    for (int off = 128; off > 0; off >>= 1) {
        if (tid < off) {
            r0[tid] += r0[tid + off]; r1[tid] += r1[tid + off];
            r2[tid] += r2[tid + off]; r3[tid] += r3[tid + off];
        }
        __syncthreads();
    }
    if (tid == 0) {
        float* cp = context + b * CC + c0;
        cp[0] = r0[0]; cp[1] = r1[0]; cp[2] = r2[0]; cp[3] = r3[0];
    }
}

// ---------------------------------------------------------------------------
// Kernel 4: channel-add MLP on the 16x256 context, fully in one wave32 using
// V_WMMA_F32_16X16X4_F32 (exact fp32 matrix math, EXEC all-ones).
//   GEMM1: t[16 batch x 16 cr] = context[16x256] * w1^T[256x16] + b1
//   LayerNorm over CR=16 (per row, via __shfl_xor within each 16-lane half,
//   matching the C/D fragment layout), ReLU
//   GEMM2: add[16 batch x 256 c] = t[16x16] * w2^T[16x256] + b2
// A fragment (16x4 f32): lanes 0-15 -> M=lane, VGPR0/1 = K+0/K+1;
//                        lanes 16-31 -> M=lane-16, VGPR0/1 = K+2/K+3.
// B fragment (4x16 f32): lanes 0-15 -> N=lane, VGPR0/1 = K+0/K+1;
//                        lanes 16-31 -> N=lane-16, VGPR0/1 = K+2/K+3.
// D (16x16 f32): VGPR r -> M=r (lanes 0-15) / M=r+8 (lanes 16-31), N=lane%16.
// ---------------------------------------------------------------------------
__global__ __launch_bounds__(32) void k_mlp_wmma(const float* __restrict__ context,
                                                 const float* __restrict__ w1,
                                                 const float* __restrict__ b1,
                                                 const float* __restrict__ ln_g,
                                                 const float* __restrict__ ln_b,
                                                 const float* __restrict__ w2,
                                                 const float* __restrict__ b2,
                                                 float* __restrict__ addv) {
    __shared__ float t_s[16 * 16];
    const int lane = threadIdx.x;
    const int half = lane >> 4;      // 0 or 1
    const int l16  = lane & 15;

    // ---- GEMM1: K = 256 in steps of 4 ----
    v8f acc = {};
    for (int k0 = 0; k0 < CC; k0 += 4) {
        const int kk = k0 + (half << 1);
        v2f a, bf;
        a.x  = context[l16 * CC + kk];
        a.y  = context[l16 * CC + kk + 1];
        bf.x = w1[l16 * CC + kk];        // B[k][n] = w1[n][k]
        bf.y = w1[l16 * CC + kk + 1];
        acc = __builtin_amdgcn_wmma_f32_16x16x4_f32(
                  false, a, false, bf, (short)0, acc, false, false);
    }

    // ---- bias + LayerNorm(CR=16) + ReLU, per output row ----
    const float bias1 = b1[l16];
    const float g = ln_g[l16], be = ln_b[l16];
    #pragma unroll
    for (int r = 0; r < 8; ++r) {
        float v = acc[r] + bias1;
        float s = v;                          // sum over 16 lanes of this half
        s += __shfl_xor(s, 1);
        s += __shfl_xor(s, 2);
        s += __shfl_xor(s, 4);
        s += __shfl_xor(s, 8);
        const float mu = s * (1.f / 16.f);
        const float d = v - mu;
        float q = d * d;
        q += __shfl_xor(q, 1);
        q += __shfl_xor(q, 2);
        q += __shfl_xor(q, 4);
        q += __shfl_xor(q, 8);
        const float var = q * (1.f / 16.f);
        float y = d * rsqrtf(var + LN_EPS) * g + be;
        y = fmaxf(y, 0.f);
        t_s[(r + (half << 3)) * 16 + l16] = y;   // t[M][N], M=batch, N=cr
    }
    __syncthreads();

    // ---- GEMM2: N = 256 in 16 tiles, K = 16 in steps of 4 ----
    for (int nt = 0; nt < 16; ++nt) {
        const int n = nt * 16 + l16;
        v8f acc2 = {};
        #pragma unroll
        for (int k0 = 0; k0 < CR; k0 += 4) {
            const int kk = k0 + (half << 1);
            v2f a, bf;
            a.x  = t_s[l16 * 16 + kk];
            a.y  = t_s[l16 * 16 + kk + 1];
            bf.x = w2[n * CR + kk];          // B[k][n] = w2[n][k]
            bf.y = w2[n * CR + kk + 1];
            acc2 = __builtin_amdgcn_wmma_f32_16x16x4_f32(
                       false, a, false, bf, (short)0, acc2, false, false);
        }
        const float bias2 = b2[n];
        #pragma unroll
        for (int r = 0; r < 8; ++r)
            addv[(r + (half << 3)) * CC + n] = acc2[r] + bias2;
    }
}

// ---------------------------------------------------------------------------
// Kernel 5: out = x + add[b][c] broadcast. Third read of x + full write,
// float4 vectorized (global_load/store_b128). bc index = element >> 14.
// ---------------------------------------------------------------------------
__global__ __launch_bounds__(256) void k_add(const float* __restrict__ x,
                                             const float* __restrict__ addv,
                                             float* __restrict__ out) {
    const size_t g = (size_t)blockIdx.x * 256 + threadIdx.x;  // float4 index
    const int bc = (int)((g * 4) >> 14);                      // S = 16384
    const float a = addv[bc];
    float4 v = ((const float4*)x)[g];
    v.x += a; v.y += a; v.z += a; v.w += a;
    ((float4*)out)[g] = v;
}

extern "C" void kernel_launch(void* const* d_in, const int* in_sizes, int n_in,
                              void* d_out, int out_size, void* d_ws, size_t ws_size,
                              hipStream_t stream) {
    const float* x      = (const float*)d_in[0];
    const float* w_mask = (const float*)d_in[1];
    const float* b_mask = (const float*)d_in[2];
    const float* w1     = (const float*)d_in[3];
    const float* b1     = (const float*)d_in[4];
    const float* ln_g   = (const float*)d_in[5];
    const float* ln_b   = (const float*)d_in[6];
    const float* w2     = (const float*)d_in[7];
    const float* b2     = (const float*)d_in[8];
    float* out = (float*)d_out;

    // workspace layout: logits/attn [B*S] | context [B*C] | add [B*C]
    float* logits  = (float*)d_ws;
    float* context = logits + BB * SS;
    float* addv    = context + BB * CC;

    k_logits <<<BB * (SS / 256), 256, 0, stream>>>(x, w_mask, b_mask, logits);
    k_softmax<<<BB, 1024, 0, stream>>>(logits);
    k_context<<<BB * (CC / 4), 256, 0, stream>>>(x, logits, context);
    k_mlp_wmma<<<1, 32, 0, stream>>>(context, w1, b1, ln_g, ln_b, w2, b2, addv);
    k_add    <<<(size_t)BB * CC * SS / (256 * 4), 256, 0, stream>>>(x, addv, out);
}